// MultiResHashGridEncoder_37898791420382
// MI455X (gfx1250) — compile-verified
//
#include <hip/hip_runtime.h>
#include <stdint.h>

// Instant-NGP multiresolution hash-grid encode for gfx1250 (MI455X).
// Gather-bound: 64MB table is L2-resident (192MB L2); optimize for gather MLP,
// async-LDS staging of coords, and NT output stores that don't evict the table.

typedef float v2f __attribute__((ext_vector_type(2)));

#define NLEVELS 16
#define LOG2T   19
#define TMASK   ((1u << LOG2T) - 1u)
#define PRIME1  2654435761u
#define PRIME2  805459861u
#define PTS     64          // points per block
#define THREADS 256         // 8 wave32 waves; lane layout: l = t&15, point-slot = t>>4

// floor(16 * 2^(l/3)) for l = 0..15 (BASE_RES=16, FINEST_RES=512, L=16)
__constant__ float c_res[NLEVELS] = {
    16.f, 20.f, 25.f, 32.f, 40.f, 50.f, 64.f, 80.f,
    101.f, 128.f, 161.f, 203.f, 256.f, 322.f, 406.f, 512.f
};

__global__ __launch_bounds__(THREADS)
void hashgrid_encode_kernel(const float* __restrict__ x,
                            const float* __restrict__ table,
                            float* __restrict__ out,
                            int n_points)
{
    __shared__ float sx[PTS * 3];          // 768 B of staged coords
    const int t  = (int)threadIdx.x;
    const int n0 = (int)blockIdx.x * PTS;

    // --- CDNA5 async copy: stage this block's 64 points (192 dwords) into LDS.
    // GVS mode: mem = SGPR_base + VGPR_offset; VDST holds LDS byte address.
    if (t < PTS * 3) {
        uint32_t lds_addr = (uint32_t)(uintptr_t)(&sx[t]);          // low 32b of flat = LDS offset
        uint32_t byte_off = ((uint32_t)(n0 * 3) + (uint32_t)t) * 4u;
        asm volatile("global_load_async_to_lds_b32 %0, %1, %2 offset:0"
                     :: "v"(lds_addr), "v"(byte_off),
                        "s"((uint64_t)(uintptr_t)x)
                     : "memory");
    }
    asm volatile("s_wait_asynccnt 0x0" ::: "memory");
    __syncthreads();

    const int   l   = t & (NLEVELS - 1);   // level handled by this lane
    const int   p0  = t >> 4;              // point slot 0..15
    const float res = c_res[l];
    const v2f* __restrict__ tl = (const v2f*)table + ((size_t)l << LOG2T); // level's table
    v2f* __restrict__ o2 = (v2f*)out;

#pragma unroll
    for (int k = 0; k < 4; ++k) {
        const int p = p0 + k * 16;
        const int n = n0 + p;
        if (n < n_points) {
            // scaled coords, integer base, fractional part
            const float X = sx[p * 3 + 0] * res;
            const float Y = sx[p * 3 + 1] * res;
            const float Z = sx[p * 3 + 2] * res;
            const float bx = floorf(X), by = floorf(Y), bz = floorf(Z);
            const float fx = X - bx, fy = Y - by, fz = Z - bz;
            const uint32_t ux = (uint32_t)bx, uy = (uint32_t)by, uz = (uint32_t)bz;

            // strength-reduced corner hashes: h = x' ^ y'*p1 ^ z'*p2
            const uint32_t ax0 = ux,            ax1 = ux + 1u;
            const uint32_t ay0 = uy * PRIME1,   ay1 = ay0 + PRIME1;
            const uint32_t az0 = uz * PRIME2,   az1 = az0 + PRIME2;
            const uint32_t yz00 = ay0 ^ az0, yz10 = ay1 ^ az0;
            const uint32_t yz01 = ay0 ^ az1, yz11 = ay1 ^ az1;

            // 8 independent b64 gathers (L2-resident table) -- issue together for MLP
            const v2f f0 = tl[(ax0 ^ yz00) & TMASK];
            const v2f f1 = tl[(ax1 ^ yz00) & TMASK];
            const v2f f2 = tl[(ax0 ^ yz10) & TMASK];
            const v2f f3 = tl[(ax1 ^ yz10) & TMASK];
            const v2f f4 = tl[(ax0 ^ yz01) & TMASK];
            const v2f f5 = tl[(ax1 ^ yz01) & TMASK];
            const v2f f6 = tl[(ax0 ^ yz11) & TMASK];
            const v2f f7 = tl[(ax1 ^ yz11) & TMASK];

            // trilinear weights (corner bit d selects frac_d vs 1-frac_d)
            const float gx = 1.f - fx, gy = 1.f - fy, gz = 1.f - fz;
            const float c00 = gy * gz, c10 = fy * gz, c01 = gy * fz, c11 = fy * fz;
            const float w0 = gx * c00, w1 = fx * c00, w2 = gx * c10, w3 = fx * c10;
            const float w4 = gx * c01, w5 = fx * c01, w6 = gx * c11, w7 = fx * c11;

            float r0 = w0 * f0.x; float r1 = w0 * f0.y;
            r0 = fmaf(w1, f1.x, r0); r1 = fmaf(w1, f1.y, r1);
            r0 = fmaf(w2, f2.x, r0); r1 = fmaf(w2, f2.y, r1);
            r0 = fmaf(w3, f3.x, r0); r1 = fmaf(w3, f3.y, r1);
            r0 = fmaf(w4, f4.x, r0); r1 = fmaf(w4, f4.y, r1);
            r0 = fmaf(w5, f5.x, r0); r1 = fmaf(w5, f5.y, r1);
            r0 = fmaf(w6, f6.x, r0); r1 = fmaf(w6, f6.y, r1);
            r0 = fmaf(w7, f7.x, r0); r1 = fmaf(w7, f7.y, r1);

            v2f acc; acc.x = r0; acc.y = r1;
            // NT store: output is write-once/stream -> don't evict the L2-resident table.
            __builtin_nontemporal_store(acc, &o2[(size_t)n * NLEVELS + l]);
        }
    }
}

extern "C" void kernel_launch(void* const* d_in, const int* in_sizes, int n_in,
                              void* d_out, int out_size, void* d_ws, size_t ws_size,
                              hipStream_t stream) {
    const float* x     = (const float*)d_in[0];   // [N,3] fp32
    const float* table = (const float*)d_in[1];   // [16, 2^19, 2] fp32
    float*       out   = (float*)d_out;           // [N, 32] fp32
    const int n_points = in_sizes[0] / 3;
    const int blocks   = (n_points + PTS - 1) / PTS;
    hipLaunchKernelGGL(hashgrid_encode_kernel, dim3(blocks), dim3(THREADS), 0, stream,
                       x, table, out, n_points);
}